// GatedProjectedInterMeanLoss_60078002536929
// MI455X (gfx1250) — compile-verified
//
#include <hip/hip_runtime.h>
#include <hip/hip_bf16.h>

#define N_ROWS 262144
#define C_CLS  128
#define D_DIM  256
#define EPSF   1e-8f

typedef float v2f __attribute__((ext_vector_type(2)));
typedef float v8f __attribute__((ext_vector_type(8)));

// ---------------------------------------------------------------------------
// Kernel 1: per-row argmax over 128 logits (one wave32 per row) + gating.
// labels[row] = argmax if gated else -1; counts[label] += 1 (f32 HW atomic).
// ---------------------------------------------------------------------------
__global__ __launch_bounds__(256) void argmax_gate_kernel(
    const float* __restrict__ logits, const int* __restrict__ gate,
    int* __restrict__ labels, float* __restrict__ counts, int n)
{
    int wave   = (blockIdx.x * blockDim.x + threadIdx.x) >> 5;
    int lane   = threadIdx.x & 31;
    int nwaves = (gridDim.x * blockDim.x) >> 5;
    for (int row = wave; row < n; row += nwaves) {
        const float* lg = logits + (size_t)row * C_CLS;
        __builtin_prefetch(lg + (size_t)nwaves * C_CLS, 0, 1);
        float best = lg[lane];
        int   bidx = lane;
        #pragma unroll
        for (int k = 1; k < 4; ++k) {           // lane owns lane, lane+32, +64, +96
            float v = lg[lane + 32 * k];
            if (v > best) { best = v; bidx = lane + 32 * k; }
        }
        #pragma unroll
        for (int off = 16; off > 0; off >>= 1) { // wave32 butterfly, first-max tiebreak
            float ov = __shfl_xor(best, off, 32);
            int   oi = __shfl_xor(bidx, off, 32);
            if (ov > best || (ov == best && oi < bidx)) { best = ov; bidx = oi; }
        }
        if (lane == 0) {
            int g = (gate[row] != 0);
            labels[row] = g ? bidx : -1;
            if (g) unsafeAtomicAdd(&counts[bidx], 1.0f);
        }
    }
}

// ---------------------------------------------------------------------------
// Kernel 2: gated segment-sum of img_feats into sums[C][D], staged in 64KB LDS
// (one D-half of 128 dims per blockIdx.y). ds_add_f32 in LDS, one global
// f32-atomic flush per block.
// ---------------------------------------------------------------------------
__global__ __launch_bounds__(256) void segsum_kernel(
    const float* __restrict__ img, const int* __restrict__ labels,
    float* __restrict__ sums, int n)
{
    __shared__ float lsum[C_CLS * 128];           // 64 KB
    int t = threadIdx.x;                          // 0..255
    for (int i = t; i < C_CLS * 128; i += 256) lsum[i] = 0.0f;
    __syncthreads();

    int half = blockIdx.y;                        // which 128-dim half of D
    int d    = t & 127;
    int sub  = t >> 7;                            // two rows in flight per block
    int rowsPerBlock = (n + gridDim.x - 1) / gridDim.x;
    int r0 = blockIdx.x * rowsPerBlock;
    int r1 = min(n, r0 + rowsPerBlock);
    for (int row = r0 + sub; row < r1; row += 2) {
        int lab = labels[row];
        __builtin_prefetch(img + (size_t)(row + 2) * D_DIM + half * 128 + d, 0, 1);
        if (lab >= 0) {
            float v = img[(size_t)row * D_DIM + half * 128 + d];
            atomicAdd(&lsum[lab * 128 + d], v);   // ds_add_f32
        }
    }
    __syncthreads();
    for (int i = t; i < C_CLS * 128; i += 256) {
        float v = lsum[i];
        if (v != 0.0f) {
            int c = i >> 7, dd = i & 127;
            unsafeAtomicAdd(&sums[c * D_DIM + half * 128 + dd], v);
        }
    }
}

// ---------------------------------------------------------------------------
// Generic f32 WMMA GEMM: C[M,N] = A[M,K] * op(B), one wave per 16x16 tile.
// V_WMMA_F32_16X16X4_F32 layouts:
//   A 16x4 : lane&15 = M row, VGPR{0,1} = K {2*(lane>>4), 2*(lane>>4)+1}
//   B 4x16 : lane&15 = N col, VGPR{0,1} = same K pair
//   C 16x16: VGPR r -> M = r + 8*(lane>>4), N = lane&15
// ---------------------------------------------------------------------------
__global__ __launch_bounds__(32) void wmma_gemm_f32(
    const float* __restrict__ A, const float* __restrict__ B, float* __restrict__ Cm,
    int K, int lda, int ldb, int ldc, int transB)
{
    int lane = threadIdx.x;
    int half = lane >> 4;
    int li   = lane & 15;
    int row0 = blockIdx.y * 16;
    int col0 = blockIdx.x * 16;

    v8f acc = {0.f, 0.f, 0.f, 0.f, 0.f, 0.f, 0.f, 0.f};
    if (transB) {
        for (int k = 0; k < K; k += 4) {
            int ka = k + 2 * half;
            v2f a, b;
            a.x = A[(row0 + li) * lda + ka];
            a.y = A[(row0 + li) * lda + ka + 1];
            b.x = B[(col0 + li) * ldb + ka];
            b.y = B[(col0 + li) * ldb + ka + 1];
            acc = __builtin_amdgcn_wmma_f32_16x16x4_f32(
                false, a, false, b, (short)0, acc, false, false);
        }
    } else {
        for (int k = 0; k < K; k += 4) {
            int ka = k + 2 * half;
            v2f a, b;
            a.x = A[(row0 + li) * lda + ka];
            a.y = A[(row0 + li) * lda + ka + 1];
            b.x = B[ka * ldb + col0 + li];
            b.y = B[(ka + 1) * ldb + col0 + li];
            acc = __builtin_amdgcn_wmma_f32_16x16x4_f32(
                false, a, false, b, (short)0, acc, false, false);
        }
    }
    #pragma unroll
    for (int r = 0; r < 8; ++r)
        Cm[(row0 + r + 8 * half) * ldc + col0 + li] = acc[r];
}

// ---------------------------------------------------------------------------
// Gauss-Jordan inversion of SPD 128x128 Gram (no pivoting needed).
// aug = [G | I] is 128x256 in global scratch; single block of 256 threads.
// ---------------------------------------------------------------------------
__global__ __launch_bounds__(256) void gj_init(const float* __restrict__ G,
                                               float* __restrict__ aug)
{
    int i = blockIdx.x * blockDim.x + threadIdx.x;  // 128*256 entries
    int r = i >> 8, c = i & 255;
    aug[i] = (c < C_CLS) ? G[r * C_CLS + c] : ((c - C_CLS) == r ? 1.0f : 0.0f);
}

__global__ __launch_bounds__(256) void gj_invert(float* __restrict__ aug)
{
    __shared__ float rowk[256];
    __shared__ float fct[C_CLS];
    int t = threadIdx.x;                             // 256 threads, t = column
    for (int k = 0; k < C_CLS; ++k) {
        float piv = aug[k * 256 + k];
        __syncthreads();                             // all read pivot before write
        float v = aug[k * 256 + t] / piv;
        aug[k * 256 + t] = v;
        rowk[t] = v;
        if (t < C_CLS) fct[t] = (t == k) ? 0.0f : aug[t * 256 + k];
        __syncthreads();
        for (int j = 0; j < C_CLS; ++j)              // eliminate column k everywhere
            aug[j * 256 + t] -= fct[j] * rowk[t];
        __syncthreads();
    }
}

// ---------------------------------------------------------------------------
// means = proj/max(cnt,1); unit = means / (safe_norm + eps); present flags.
// One block of 256 threads (one per dim) per class.
// ---------------------------------------------------------------------------
__global__ __launch_bounds__(256) void normalize_kernel(
    const float* __restrict__ proj, const float* __restrict__ counts,
    float* __restrict__ unitv, float* __restrict__ present)
{
    __shared__ float red[256];
    int c = blockIdx.x, t = threadIdx.x;
    float cnt = counts[c];
    float m = proj[c * D_DIM + t] / fmaxf(cnt, 1.0f);
    red[t] = m * m;
    __syncthreads();
    for (int s = 128; s > 0; s >>= 1) { if (t < s) red[t] += red[t + s]; __syncthreads(); }
    bool pres = cnt > 0.0f;
    float norm = sqrtf(pres ? red[0] : 1.0f);
    unitv[c * D_DIM + t] = m / (norm + EPSF);
    if (t == 0) present[c] = pres ? 1.0f : 0.0f;
}

// ---------------------------------------------------------------------------
// loss = sum over off-diagonal present pairs of (1 - sim); 0 if n_present < 2.
// ---------------------------------------------------------------------------
__global__ __launch_bounds__(256) void loss_kernel(
    const float* __restrict__ sim, const float* __restrict__ present,
    float* __restrict__ out)
{
    __shared__ float red[256];
    __shared__ float pcnt[256];
    int t = threadIdx.x;
    float acc = 0.0f, np = 0.0f;
    for (int i = t; i < C_CLS * C_CLS; i += 256) {
        int a = i >> 7, b = i & 127;
        if (a != b && present[a] > 0.0f && present[b] > 0.0f) acc += 1.0f - sim[i];
    }
    for (int c = t; c < C_CLS; c += 256) np += present[c];
    red[t] = acc; pcnt[t] = np;
    __syncthreads();
    for (int s = 128; s > 0; s >>= 1) {
        if (t < s) { red[t] += red[t + s]; pcnt[t] += pcnt[t + s]; }
        __syncthreads();
    }
    if (t == 0) out[0] = (pcnt[0] >= 2.0f) ? red[0] : 0.0f;
}

// ---------------------------------------------------------------------------
extern "C" void kernel_launch(void* const* d_in, const int* in_sizes, int n_in,
                              void* d_out, int out_size, void* d_ws, size_t ws_size,
                              hipStream_t stream)
{
    const float* logits = (const float*)d_in[0];   // (N, C)
    const float* img    = (const float*)d_in[1];   // (N, D)
    const float* text   = (const float*)d_in[2];   // (C, D), L2-normalized rows
    const int*   gate   = (const int*)d_in[3];     // (N,)
    float* out = (float*)d_out;

    // workspace layout (floats)
    float* w       = (float*)d_ws;
    float* sums    = w;                  // C*D   = 32768
    float* counts  = sums + 32768;       // C     = 128
    float* G       = counts + 128;       // C*C   = 16384
    float* aug     = G + 16384;          // C*256 = 32768
    float* A1      = aug + 32768;        // C*C
    float* A2      = A1 + 16384;         // C*C
    float* proj    = A2 + 16384;         // C*D
    float* unitv   = proj + 32768;       // C*D
    float* present = unitv + 32768;      // C
    float* simm    = present + 128;      // C*C
    int*   labels  = (int*)(simm + 16384); // N ints

    hipMemsetAsync(sums, 0, (size_t)(32768 + 128) * sizeof(float), stream);

    // Phase 1: argmax + gate, then LDS-staged gated segment sum (bandwidth-bound)
    argmax_gate_kernel<<<1024, 256, 0, stream>>>(logits, gate, labels, counts, N_ROWS);
    segsum_kernel<<<dim3(512, 2), 256, 0, stream>>>(img, labels, sums, N_ROWS);

    // Phase 2: projector math on class sums via f32 WMMA GEMMs
    // G = T * T^T            (128x128, K=256)
    wmma_gemm_f32<<<dim3(8, 8), 32, 0, stream>>>(text, text, G, 256, 256, 256, 128, 1);
    gj_init<<<128, 256, 0, stream>>>(G, aug);
    gj_invert<<<1, 256, 0, stream>>>(aug);
    // A1 = sums * T^T        (128x128, K=256)
    wmma_gemm_f32<<<dim3(8, 8), 32, 0, stream>>>(sums, text, A1, 256, 256, 256, 128, 1);
    // A2 = A1 * Ginv         (128x128, K=128), Ginv lives at aug+128 with ld=256
    wmma_gemm_f32<<<dim3(8, 8), 32, 0, stream>>>(A1, aug + 128, A2, 128, 128, 256, 128, 0);
    // proj = A2 * T          (128x256, K=128)  == projected class sums
    wmma_gemm_f32<<<dim3(16, 8), 32, 0, stream>>>(A2, text, proj, 128, 128, 256, 256, 0);

    // Phase 3: normalize means, cosine Gram via WMMA, masked pair loss
    normalize_kernel<<<128, 256, 0, stream>>>(proj, counts, unitv, present);
    // sim = unit * unit^T    (128x128, K=256)
    wmma_gemm_f32<<<dim3(8, 8), 32, 0, stream>>>(unitv, unitv, simm, 256, 256, 256, 128, 1);
    loss_kernel<<<1, 256, 0, stream>>>(simm, present, out);
}